// MultiScaleAttention_3607772529420
// MI455X (gfx1250) — compile-verified
//
#include <hip/hip_runtime.h>

#define DEV_ __device__ __forceinline__

typedef float v2f __attribute__((ext_vector_type(2)));
typedef float v8f __attribute__((ext_vector_type(8)));

// Problem constants
constexpr int Bc = 8, NFc = 384, Dc = 512, Hc = 16, Kc = 4, DKc = 32;
constexpr int HKc = Hc * Kc;                       // 64
constexpr float SCALE_QK = 0.17677669529663687f;   // 1/sqrt(32)

// ---------------------------------------------------------------------------
// Generic WMMA f32 16x16xK accumulation loop (V_WMMA_F32_16X16X4_F32).
// A element [m][kk] at Ab + m*ars + kk           (row-major A tile, 16 x 4*ksteps)
// B element [kk][n] at Bb + kk*brs + n*bcs       (arbitrary strides: handles B^T)
// Fragment layout per ISA 7.12.2:
//   A: lanes 0-15 m=lane, v[0..1]=K{0,1}; lanes 16-31 v[0..1]=K{2,3}
//   B: lanes 0-15 n=lane, v[0..1]=K{0,1}; lanes 16-31 v[0..1]=K{2,3}
//   C/D: VGPR r -> row (lane<16 ? r : 8+r), col lane&15
// ---------------------------------------------------------------------------
DEV_ v8f wmma_loop(const float* __restrict__ Ab, int ars,
                   const float* __restrict__ Bb, int brs, int bcs,
                   int ksteps, v8f acc)
{
  const int lane = threadIdx.x & 31;
  const int half = lane >> 4;
  const int l15  = lane & 15;
  const float* arow = Ab + (size_t)l15 * ars + half * 2;
  const float* bcol = Bb + (size_t)l15 * bcs + (size_t)(half * 2) * brs;
  for (int kk = 0; kk < ksteps; ++kk) {
    v2f a, b;
    a.x = arow[0];
    a.y = arow[1];
    b.x = bcol[0];
    b.y = bcol[brs];
    acc = __builtin_amdgcn_wmma_f32_16x16x4_f32(false, a, false, b,
                                                (short)0, acc, false, false);
    arow += 4;
    bcol += (size_t)4 * brs;
  }
  return acc;
}

// ---------------------------------------------------------------------------
// Generic tiled GEMM: C[M,N] = act(A[M,K] @ W[K,N] + bias) (+resid). One 16x16
// tile per wave32. act: 0=none, 1=SiLU.
// ---------------------------------------------------------------------------
__global__ void k_gemm(const float* __restrict__ A, int lda,
                       const float* __restrict__ W, int ldw,
                       const float* __restrict__ bias,
                       const float* __restrict__ resid, int ldr,
                       float* __restrict__ C, int ldc,
                       int tilesM, int tilesN, int ksteps, int act)
{
  int wave = (int)((blockIdx.x * blockDim.x + threadIdx.x) >> 5);
  if (wave >= tilesM * tilesN) return;
  int tm = wave / tilesN, tn = wave % tilesN;
  v8f acc = {};
  acc = wmma_loop(A + (size_t)(tm * 16) * lda, lda, W + tn * 16, ldw, 1, ksteps, acc);
  int lane = threadIdx.x & 31, half = lane >> 4, l15 = lane & 15;
  int col = tn * 16 + l15;
  float bv = bias ? bias[col] : 0.0f;
#pragma unroll
  for (int r = 0; r < 8; ++r) {
    int row = tm * 16 + half * 8 + r;
    float v = acc[r] + bv;
    if (act == 1) v = v / (1.0f + __expf(-v));   // SiLU
    if (resid) v += resid[(size_t)row * ldr + col];
    C[(size_t)row * ldc + col] = v;
  }
}

// ---------------------------------------------------------------------------
// S[b,h,i,j] = (q[b,i,h,:] . k[b,j,h,:]) / sqrt(dk).  B fragment loaded
// transposed via strides (brs=1, bcs=Dc).
// ---------------------------------------------------------------------------
__global__ void k_scores(const float* __restrict__ q, const float* __restrict__ k,
                         float* __restrict__ S)
{
  int wave = (int)((blockIdx.x * blockDim.x + threadIdx.x) >> 5);
  constexpr int TI = NFc / 16;       // 24
  constexpr int TILES = TI * TI;     // 576
  if (wave >= Bc * Hc * TILES) return;
  int bh = wave / TILES, t = wave % TILES;
  int b = bh / Hc, h = bh % Hc;
  int ti = t / TI, tj = t % TI;
  const float* qb = q + (size_t)(b * NFc + ti * 16) * Dc + h * DKc;
  const float* kb = k + (size_t)(b * NFc + tj * 16) * Dc + h * DKc;
  v8f acc = {};
  acc = wmma_loop(qb, Dc, kb, /*brs=*/1, /*bcs=*/Dc, DKc / 4, acc);
  float* Sb = S + (size_t)bh * NFc * NFc + (size_t)(ti * 16) * NFc + tj * 16;
  int lane = threadIdx.x & 31, half = lane >> 4, l15 = lane & 15;
#pragma unroll
  for (int r = 0; r < 8; ++r)
    Sb[(size_t)(half * 8 + r) * NFc + l15] = acc[r] * SCALE_QK;
}

// ---------------------------------------------------------------------------
// Block reduction helpers (wave32 shfl + LDS across waves).
// ---------------------------------------------------------------------------
DEV_ float block_reduce(float v, bool domax, float* red, float* bc)
{
#pragma unroll
  for (int o = 16; o > 0; o >>= 1) {
    float other = __shfl_xor(v, o, 32);
    v = domax ? fmaxf(v, other) : (v + other);
  }
  int wid = threadIdx.x >> 5, lane = threadIdx.x & 31;
  __syncthreads();
  if (lane == 0) red[wid] = v;
  __syncthreads();
  if (threadIdx.x == 0) {
    float r = red[0];
    int nw = blockDim.x >> 5;
    for (int w = 1; w < nw; ++w) r = domax ? fmaxf(r, red[w]) : (r + red[w]);
    *bc = r;
  }
  __syncthreads();
  return *bc;
}

// ---------------------------------------------------------------------------
// Softmax over j with per-scale distance masking. One block per (b,h,i) row,
// 384 threads (one per j); S/bias/dist read once, 4 attn rows written.
// ---------------------------------------------------------------------------
__global__ void k_softmax(const float* __restrict__ S, const float* __restrict__ bias,
                          const float* __restrict__ dist, const float* __restrict__ dbar,
                          float* __restrict__ attn)
{
  __shared__ float red[16];
  __shared__ float bc;
  int j = threadIdx.x;
  int rowid = blockIdx.x;          // bh*NFc + i
  int i = rowid % NFc;
  int bh = rowid / NFc;
  int b = bh / Hc;
  float s = S[(size_t)rowid * NFc + j] + bias[((size_t)b * NFc + i) * NFc + j];
  float dval = (i == 0 || j == 0)
                   ? 0.0f
                   : dist[((size_t)b * (NFc - 1) + (i - 1)) * (NFc - 1) + (j - 1)];
  for (int ks = 0; ks < Kc; ++ks) {
    float val = (dval < dbar[ks]) ? s : -1e12f;
    float mx = block_reduce(val, true, red, &bc);
    float e = __expf(val - mx);
    float sum = block_reduce(e, false, red, &bc);
    attn[((size_t)(bh * Kc + ks) * NFc + i) * NFc + j] = e / sum;
  }
}

// ---------------------------------------------------------------------------
// v_msg[b,h,k] = attn[b,h,k] (384x384) @ v[b,:,h,:] (384x32)
// ---------------------------------------------------------------------------
__global__ void k_vmsg(const float* __restrict__ attn, const float* __restrict__ v,
                       float* __restrict__ vmsg)
{
  int wave = (int)((blockIdx.x * blockDim.x + threadIdx.x) >> 5);
  constexpr int TI = NFc / 16;        // 24
  constexpr int TD = DKc / 16;        // 2
  constexpr int TILES = TI * TD;      // 48
  if (wave >= Bc * HKc * TILES) return;
  int bhk = wave / TILES, t = wave % TILES;
  int ti = t / TD, td = t % TD;
  int b = bhk / HKc;
  int h = (bhk % HKc) / Kc;
  const float* Ab = attn + (size_t)bhk * NFc * NFc + (size_t)(ti * 16) * NFc;
  const float* Bb = v + (size_t)(b * NFc) * Dc + h * DKc + td * 16;
  v8f acc = {};
  acc = wmma_loop(Ab, NFc, Bb, /*brs=*/Dc, /*bcs=*/1, NFc / 4, acc);
  float* Cb = vmsg + (size_t)bhk * NFc * DKc + (size_t)(ti * 16) * DKc + td * 16;
  int lane = threadIdx.x & 31, half = lane >> 4, l15 = lane & 15;
#pragma unroll
  for (int r = 0; r < 8; ++r)
    Cb[(size_t)(half * 8 + r) * DKc + l15] = acc[r];
}

// ---------------------------------------------------------------------------
// dist_msg: per (b,i):  attn[b,:,:,i,:] (64x384, row stride N*N) @ de[b,i] (384x32).
// Streams dist_embed exactly once.
// ---------------------------------------------------------------------------
__global__ void k_dmsg(const float* __restrict__ attn, const float* __restrict__ de,
                       float* __restrict__ dmsg)
{
  int wave = (int)((blockIdx.x * blockDim.x + threadIdx.x) >> 5);
  constexpr int TR = HKc / 16;        // 4
  constexpr int TD = DKc / 16;        // 2
  constexpr int TILES = TR * TD;      // 8
  if (wave >= Bc * NFc * TILES) return;
  int bi = wave / TILES, t = wave % TILES;
  int tr = t / TD, td = t % TD;
  int b = bi / NFc, i = bi % NFc;
  const float* Ab = attn + (size_t)b * HKc * NFc * NFc
                         + (size_t)(tr * 16) * NFc * NFc + (size_t)i * NFc;
  const float* Bb = de + (size_t)(b * NFc + i) * NFc * DKc + td * 16;
  v8f acc = {};
  acc = wmma_loop(Ab, NFc * NFc, Bb, /*brs=*/DKc, /*bcs=*/1, NFc / 4, acc);
  float* Cb = dmsg + ((size_t)(b * HKc + tr * 16) * NFc + i) * DKc + td * 16;
  int lane = threadIdx.x & 31, half = lane >> 4, l15 = lane & 15;
#pragma unroll
  for (int r = 0; r < 8; ++r)
    Cb[(size_t)(half * 8 + r) * (NFc * DKc) + l15] = acc[r];
}

// ---------------------------------------------------------------------------
// msg = v_msg + dist_msg, permuted into the reference's flat m layout:
// m[b][(k*H+h)*dk + d][n]  (per-batch flat [K*H*dk, N]).
// ---------------------------------------------------------------------------
__global__ void k_pack(const float* __restrict__ vmsg, const float* __restrict__ dmsg,
                       float* __restrict__ m)
{
  size_t idx = (size_t)blockIdx.x * blockDim.x + threadIdx.x;
  constexpr size_t TOT = (size_t)Bc * Hc * Kc * NFc * DKc;
  if (idx >= TOT) return;
  int d = (int)(idx % DKc);  size_t t = idx / DKc;
  int n = (int)(t % NFc);    t /= NFc;
  int ks = (int)(t % Kc);    t /= Kc;
  int h = (int)(t % Hc);
  int b = (int)(t / Hc);
  float val = vmsg[idx]
            + dmsg[((size_t)(b * HKc + h * Kc + ks) * NFc + n) * DKc + d];
  m[(size_t)b * NFc * (Kc * Dc) + (size_t)((ks * Hc + h) * DKc + d) * NFc + n] = val;
}

// ---------------------------------------------------------------------------
// LayerNorm: one block per token row (512 threads, one per channel).
// ---------------------------------------------------------------------------
__global__ void k_ln(const float* __restrict__ y, const float* __restrict__ g,
                     const float* __restrict__ be, float* __restrict__ out)
{
  __shared__ float red[16];
  __shared__ float bc;
  int row = blockIdx.x;
  int t = threadIdx.x;
  float v = y[(size_t)row * Dc + t];
  float mu = block_reduce(v, false, red, &bc) * (1.0f / Dc);
  float dv = v - mu;
  float var = block_reduce(dv * dv, false, red, &bc) * (1.0f / Dc);
  out[(size_t)row * Dc + t] = dv * rsqrtf(var + 1e-6f) * g[t] + be[t];
}

// ---------------------------------------------------------------------------
extern "C" void kernel_launch(void* const* d_in, const int* in_sizes, int n_in,
                              void* d_out, int out_size, void* d_ws, size_t ws_size,
                              hipStream_t stream)
{
  const float* x     = (const float*)d_in[0];
  const float* dist  = (const float*)d_in[1];
  const float* dbar  = (const float*)d_in[2];
  const float* abias = (const float*)d_in[3];
  const float* de    = (const float*)d_in[4];
  const float* Wq = (const float*)d_in[5];   const float* bq = (const float*)d_in[6];
  const float* Wk = (const float*)d_in[7];   const float* bk = (const float*)d_in[8];
  const float* Wv = (const float*)d_in[9];   const float* bv = (const float*)d_in[10];
  const float* W1 = (const float*)d_in[11];  const float* b1 = (const float*)d_in[12];
  const float* W2 = (const float*)d_in[13];  const float* b2 = (const float*)d_in[14];
  const float* lng = (const float*)d_in[15]; const float* lnb = (const float*)d_in[16];

  float* out  = (float*)d_out;
  float* attn = out + (size_t)Bc * NFc * Dc;   // outputs concatenated: (out, attn)

  float* ws = (float*)d_ws;
  size_t off = 0;
  float* q    = ws + off; off += (size_t)Bc * NFc * Dc;          // 1.57M
  float* kbuf = ws + off; off += (size_t)Bc * NFc * Dc;
  float* vbuf = ws + off; off += (size_t)Bc * NFc * Dc;
  float* S    = ws + off; off += (size_t)Bc * Hc * NFc * NFc;    // 18.9M
  float* vmsg = ws + off; off += (size_t)Bc * HKc * NFc * DKc;   // 6.3M
  float* dmsg = ws + off; off += (size_t)Bc * HKc * NFc * DKc;   // 6.3M
  float* m    = ws + off; off += (size_t)Bc * NFc * (Kc * Dc);   // 6.3M
  float* t1   = ws + off; off += (size_t)Bc * NFc * Dc;
  float* y    = ws + off; off += (size_t)Bc * NFc * Dc;
  (void)off; (void)ws_size; (void)in_sizes; (void)n_in; (void)out_size;

  const int THREADS = 256;                       // 8 waves per block
  auto blocksFor = [](int waves) { return (waves + 7) / 8; };

  // 1) QKV projections: [3072,512] @ [512,512]
  {
    int tilesM = (Bc * NFc) / 16, tilesN = Dc / 16;   // 192 x 32
    int blocks = blocksFor(tilesM * tilesN);
    k_gemm<<<blocks, THREADS, 0, stream>>>(x, Dc, Wq, Dc, bq, nullptr, 0, q,    Dc, tilesM, tilesN, Dc / 4, 0);
    k_gemm<<<blocks, THREADS, 0, stream>>>(x, Dc, Wk, Dc, bk, nullptr, 0, kbuf, Dc, tilesM, tilesN, Dc / 4, 0);
    k_gemm<<<blocks, THREADS, 0, stream>>>(x, Dc, Wv, Dc, bv, nullptr, 0, vbuf, Dc, tilesM, tilesN, Dc / 4, 0);
  }

  // 2) scores = q k^T / sqrt(dk)
  k_scores<<<blocksFor(Bc * Hc * (NFc / 16) * (NFc / 16)), THREADS, 0, stream>>>(q, kbuf, S);

  // 3) masked multi-scale softmax -> attn output
  k_softmax<<<Bc * Hc * NFc, NFc, 0, stream>>>(S, abias, dist, dbar, attn);

  // 4) v_msg = attn @ v
  k_vmsg<<<blocksFor(Bc * HKc * (NFc / 16) * (DKc / 16)), THREADS, 0, stream>>>(attn, vbuf, vmsg);

  // 5) dist_msg = attn-slice @ dist_embed
  k_dmsg<<<blocksFor(Bc * NFc * (HKc / 16) * (DKc / 16)), THREADS, 0, stream>>>(attn, de, dmsg);

  // 6) pack msg into reference m layout
  {
    size_t tot = (size_t)Bc * Hc * Kc * NFc * DKc;
    k_pack<<<(int)((tot + THREADS - 1) / THREADS), THREADS, 0, stream>>>(vmsg, dmsg, m);
  }

  // 7) t1 = silu(m @ W1 + b1): [3072,2048] @ [2048,512]
  k_gemm<<<blocksFor(192 * 32), THREADS, 0, stream>>>(m, Kc * Dc, W1, Dc, b1, nullptr, 0, t1, Dc, 192, 32, (Kc * Dc) / 4, 1);

  // 8) y = t1 @ W2 + b2 + x
  k_gemm<<<blocksFor(192 * 32), THREADS, 0, stream>>>(t1, Dc, W2, Dc, b2, x, Dc, y, Dc, 192, 32, Dc / 4, 0);

  // 9) layernorm -> out
  k_ln<<<Bc * NFc, Dc, 0, stream>>>(y, lng, lnb, out);
}